// LinearAttention_62560493633831
// MI455X (gfx1250) — compile-verified
//
#include <hip/hip_runtime.h>
#include <hip/hip_bf16.h>

// ---------------------------------------------------------------------------
// Linear attention (chunked scan) for MI455X / gfx1250, fp32 WMMA path.
//   q = xWq, k = xWk, v = xWv                      (GEMM, V_WMMA_F32_16X16X4_F32)
//   M_t = M_{t-1} + v_t k_t^T ; y_t = M_t q_t      (chunked: C=16 timesteps)
//   out = y Wo                                     (GEMM)
// Scan kernel stages K chunks LDS-side with global_load_async_to_lds_b128
// (double-buffered, ASYNCcnt-tracked) and keeps the V-slice B-operands in
// registers across both WMMA contractions that consume them.
// ---------------------------------------------------------------------------

#define DIM   512
#define BATCH 4
#define SEQ   4096
#define NROWS (BATCH * SEQ)   // 16384
#define CHUNK 16
#define KSTRIDE 516           // padded LDS row (floats): bank-conflict-free,
                              // 516*4 bytes is a multiple of 16B for B128 DMA

typedef __attribute__((ext_vector_type(2))) float v2f;
typedef __attribute__((ext_vector_type(8))) float v8f;

__device__ __forceinline__ v8f wmma_f32(v2f a, v2f b, v8f c) {
  // D(16x16,f32) = A(16x4,f32) * B(4x16,f32) + C
  return __builtin_amdgcn_wmma_f32_16x16x4_f32(
      /*neg_a=*/false, a, /*neg_b=*/false, b,
      /*c_mod=*/(short)0, c, /*reuse_a=*/false, /*reuse_b=*/false);
}

// ---------------------------------------------------------------------------
// GEMM: O[z] = X (nrows x 512) * W[z] (512 x 512).
// One wave computes a 16x64 output strip (4 accumulators sharing the A tile).
// grid = (nrows/16, 512/64, nz), block = 32 (one wave).
// ---------------------------------------------------------------------------
__global__ __launch_bounds__(32) void gemm512_kernel(
    const float* __restrict__ X,
    const float* __restrict__ W0, const float* __restrict__ W1,
    const float* __restrict__ W2,
    float* __restrict__ O0, float* __restrict__ O1, float* __restrict__ O2) {
  const int lane = threadIdx.x;          // 0..31
  const int m    = lane & 15;            // row/col within 16-tile
  const int koff = (lane >> 4) << 1;     // 0 or 2 (fp32 WMMA K sub-lane split)
  const int rh   = (lane >> 4) << 3;     // +8 rows for upper half-wave in C/D

  const int row0 = blockIdx.x * 16;
  const int col0 = blockIdx.y * 64;
  const float* __restrict__ W = (blockIdx.z == 0) ? W0 : (blockIdx.z == 1) ? W1 : W2;
  float* __restrict__       O = (blockIdx.z == 0) ? O0 : (blockIdx.z == 1) ? O1 : O2;

  v8f acc0 = {}, acc1 = {}, acc2 = {}, acc3 = {};

  const float* __restrict__ xr = X + (size_t)(row0 + m) * DIM;

  for (int k = 0; k < DIM; k += 4) {
    v2f a;
    a.x = xr[k + koff];
    a.y = xr[k + koff + 1];
    const float* __restrict__ wr0 = W + (size_t)(k + koff) * DIM;
    const float* __restrict__ wr1 = wr0 + DIM;
    v2f b;
    b.x = wr0[col0 + m];        b.y = wr1[col0 + m];
    acc0 = wmma_f32(a, b, acc0);
    b.x = wr0[col0 + 16 + m];   b.y = wr1[col0 + 16 + m];
    acc1 = wmma_f32(a, b, acc1);
    b.x = wr0[col0 + 32 + m];   b.y = wr1[col0 + 32 + m];
    acc2 = wmma_f32(a, b, acc2);
    b.x = wr0[col0 + 48 + m];   b.y = wr1[col0 + 48 + m];
    acc3 = wmma_f32(a, b, acc3);
  }

#pragma unroll
  for (int i = 0; i < 8; ++i) {
    float* __restrict__ orow = O + (size_t)(row0 + rh + i) * DIM + col0 + m;
    orow[0]  = acc0[i];
    orow[16] = acc1[i];
    orow[32] = acc2[i];
    orow[48] = acc3[i];
  }
}

// ---------------------------------------------------------------------------
// Chunked linear-attention scan. One wave owns (batch b, 16-wide v-slice) and
// walks S=4096 in chunks of 16. State M^T (512 K-rows x 16 V-cols) in LDS.
// K chunks are staged into double-buffered LDS via async DMA.
// grid = (BATCH, DIM/16), block = 32.
// ---------------------------------------------------------------------------
__global__ __launch_bounds__(32) void linattn_scan_kernel(
    const float* __restrict__ q, const float* __restrict__ k,
    const float* __restrict__ v, float* __restrict__ y) {
  __shared__ float Mt[DIM][CHUNK];              // M^T: [k][v]       (32 KB)
  __shared__ float Klds[2][CHUNK * KSTRIDE];    // K chunk x2        (66 KB)
  __shared__ float Satt[CHUNK][CHUNK];          // masked scores     (1 KB)

  const int lane = threadIdx.x;
  const int m    = lane & 15;
  const int koff = (lane >> 4) << 1;
  const int rh   = (lane >> 4) << 3;

  const int b  = blockIdx.x;            // batch
  const int vs = blockIdx.y * 16;       // v-slice base column

  // zero state
  for (int i = lane; i < DIM * CHUNK; i += 32) (&Mt[0][0])[i] = 0.0f;
  __syncthreads();

  const float* __restrict__ qb = q + (size_t)b * SEQ * DIM;
  const float* __restrict__ kb = k + (size_t)b * SEQ * DIM;
  const float* __restrict__ vb = v + (size_t)b * SEQ * DIM;
  float* __restrict__       yb = y + (size_t)b * SEQ * DIM;

  // Async-stage one 16x512 K chunk into LDS buffer `buf` (B128 per lane).
  // GV mode: per-lane LDS byte offset (VDST) + 64-bit global address (VADDR).
  const unsigned int klds_base = (unsigned int)(size_t)(&Klds[0][0]);
  auto stage_k = [&](int buf, int t0n) {
    const float* __restrict__ src = kb + (size_t)t0n * DIM;
    const unsigned int dst = klds_base + (unsigned int)buf * (CHUNK * KSTRIDE * 4);
    // 16 rows * 128 float4 each = 2048 float4s; 32 lanes -> 64 async B128 ops
    for (int i = lane; i < CHUNK * (DIM / 4); i += 32) {
      const int row = i >> 7;            // /128
      const int c4  = i & 127;
      unsigned long long ga =
          (unsigned long long)(size_t)(src + (size_t)row * DIM + 4 * c4);
      unsigned int la = dst + (unsigned int)(row * KSTRIDE + 4 * c4) * 4u;
      asm volatile("global_load_async_to_lds_b128 %0, %1, off"
                   :: "v"(la), "v"(ga) : "memory");
    }
  };

  stage_k(0, 0);
  asm volatile("s_wait_asynccnt 0x0" ::: "memory");
  __syncthreads();
  int cur = 0;

  for (int t0 = 0; t0 < SEQ; t0 += CHUNK) {
    // kick off async DMA of the next K chunk into the other buffer
    if (t0 + CHUNK < SEQ) stage_k(cur ^ 1, t0 + CHUNK);

    const float* __restrict__ qrow = qb + (size_t)(t0 + m) * DIM;
    const float* __restrict__ kl   = &Klds[cur][0];
    if (t0 + CHUNK < SEQ)  // hint next Q chunk into cache (global_prefetch_b8)
      __builtin_prefetch(qrow + CHUNK * DIM, 0, 1);

    // V-slice B-operands: identical registers serve Y_intra AND all 32 kt
    // tiles of the state update -> load the 16x16 slice exactly once.
    v2f bv[4];
#pragma unroll
    for (int s4 = 0; s4 < 4; ++s4) {
      bv[s4].x = vb[(size_t)(t0 + 4 * s4 + koff) * DIM + vs + m];
      bv[s4].y = vb[(size_t)(t0 + 4 * s4 + koff + 1) * DIM + vs + m];
    }

    // ---- Y_inter(16x16) = Q_chunk(16x512) * M^T(512x16)  and
    // ---- S(16x16)      = Q_chunk * K_chunk^T  (contracted over D) ----
    v8f acc = {};
    v8f s   = {};
    for (int kk = 0; kk < DIM; kk += 4) {
      v2f a;
      a.x = qrow[kk + koff];
      a.y = qrow[kk + koff + 1];
      v2f bm;                               // B[kk][v] = M^T[kk][v]
      bm.x = Mt[kk + koff][m];
      bm.y = Mt[kk + koff + 1][m];
      acc = wmma_f32(a, bm, acc);
      v2f bk;                               // B[kk][s] = K_chunk[s][kk] (LDS)
      bk.x = kl[m * KSTRIDE + kk + koff];
      bk.y = kl[m * KSTRIDE + kk + koff + 1];
      s = wmma_f32(a, bk, s);
    }

    // causal mask (s <= t inclusive), route through LDS to A-layout
#pragma unroll
    for (int i = 0; i < 8; ++i) {
      const int trow = rh + i;
      Satt[trow][m] = (m <= trow) ? s[i] : 0.0f;
    }
    __syncthreads();

    // ---- Y_intra: acc += Satt(16x16) * V_slice(16x16) ----
#pragma unroll
    for (int s4 = 0; s4 < 4; ++s4) {
      v2f a;                                // A[t][ss] = Satt[t][ss]
      a.x = Satt[m][4 * s4 + koff];
      a.y = Satt[m][4 * s4 + koff + 1];
      acc = wmma_f32(a, bv[s4], acc);
    }

    // store y chunk for our v-slice
#pragma unroll
    for (int i = 0; i < 8; ++i)
      yb[(size_t)(t0 + rh + i) * DIM + vs + m] = acc[i];

    // ---- state update: M^T += K_chunk^T(512x16) * V_slice(16x16) ----
    for (int kt = 0; kt < DIM; kt += 16) {
      v8f c;
#pragma unroll
      for (int i = 0; i < 8; ++i) c[i] = Mt[kt + rh + i][m];
#pragma unroll
      for (int s4 = 0; s4 < 4; ++s4) {
        v2f a;                              // A[krow][t] = K_chunk[t][kt+krow]
        a.x = kl[(4 * s4 + koff) * KSTRIDE + kt + m];
        a.y = kl[(4 * s4 + koff + 1) * KSTRIDE + kt + m];
        c = wmma_f32(a, bv[s4], c);
      }
#pragma unroll
      for (int i = 0; i < 8; ++i) Mt[kt + rh + i][m] = c[i];
    }

    // next chunk's K must be resident before we flip buffers
    asm volatile("s_wait_asynccnt 0x0" ::: "memory");
    __syncthreads();
    cur ^= 1;
  }
}

// ---------------------------------------------------------------------------
extern "C" void kernel_launch(void* const* d_in, const int* in_sizes, int n_in,
                              void* d_out, int out_size, void* d_ws, size_t ws_size,
                              hipStream_t stream) {
  const float* x  = (const float*)d_in[0];
  const float* Wq = (const float*)d_in[1];
  const float* Wk = (const float*)d_in[2];
  const float* Wv = (const float*)d_in[3];
  const float* Wo = (const float*)d_in[4];
  float* out = (float*)d_out;

  const size_t nd = (size_t)NROWS * DIM;
  float* qbuf = (float*)d_ws;
  float* kbuf = qbuf + nd;
  float* vbuf = kbuf + nd;
  float* ybuf = vbuf + nd;

  // 1) fused QKV projections: three GEMMs share the X tiles via L2
  gemm512_kernel<<<dim3(NROWS / 16, DIM / 64, 3), 32, 0, stream>>>(
      x, Wq, Wk, Wv, qbuf, kbuf, vbuf);

  // 2) chunked causal linear-attention scan (async K staging)
  linattn_scan_kernel<<<dim3(BATCH, DIM / 16), 32, 0, stream>>>(
      qbuf, kbuf, vbuf, ybuf);

  // 3) output projection
  gemm512_kernel<<<dim3(NROWS / 16, DIM / 64, 1), 32, 0, stream>>>(
      ybuf, Wo, Wo, Wo, out, out, out);
}